// TemporalSoINetwork_75806172774573
// MI455X (gfx1250) — compile-verified
//
#include <hip/hip_runtime.h>
#include <hip/hip_bf16.h>
#include <stdint.h>

typedef __attribute__((ext_vector_type(2))) float v2f;
typedef __attribute__((ext_vector_type(8))) float v8f;

#define Fq    4
#define Bq    16
#define Sq    1024
#define Tq    1024
#define FEATq 256
#define Cq    20
#define REPq  64
#define KEASY 819   // (F*S)/5
#define NDROP 205   // 1024 - 819

// s_wait_asynccnt takes an immediate only -> compile-time G via templates.
template <int N>
__device__ __forceinline__ void wait_async() {
    if constexpr (N == 0)       asm volatile("s_wait_asynccnt 0"  ::: "memory");
    else if constexpr (N == 3)  asm volatile("s_wait_asynccnt 3"  ::: "memory");
    else if constexpr (N == 5)  asm volatile("s_wait_asynccnt 5"  ::: "memory");
    else if constexpr (N == 9)  asm volatile("s_wait_asynccnt 9"  ::: "memory");
    else if constexpr (N == 17) asm volatile("s_wait_asynccnt 17" ::: "memory");
    else                        asm volatile("s_wait_asynccnt 0"  ::: "memory");
}

// Issue one chunk (64*G dwords, contiguous) as G coalesced async b64 loads:
// each issue moves 64 dwords (2 per lane) global -> LDS, tracked by ASYNCcnt.
template <int G>
__device__ __forceinline__ void issue_chunk(const float* __restrict__ g,
                                            int kbase_dw, uint32_t lds_byte_base,
                                            int lane) {
#pragma unroll
    for (int it = 0; it < G; ++it) {
        const int idx = it * 64 + lane * 2;                 // even dword index
        const uint32_t ldsa = lds_byte_base + (uint32_t)idx * 4u;
        const uint32_t voff = (uint32_t)(kbase_dw + idx) * 4u;
        asm volatile("global_load_async_to_lds_b64 %0, %1, %2 offset:0"
                     :: "v"(ldsa), "v"(voff), "s"(g)
                     : "memory");
    }
}

// ---------------------------------------------------------------------------
// Fused: async-staged gather + SoI max-pool + conv(256->64,relu) + fc1(64->64)
//        + fc2(64->20,relu) via V_WMMA_F32_16X16X4_F32.
// Templated on anchor index F so L,G are constexpr (async wait immediates).
// One block = 16 proposals of one (f,b). 128 threads = 4 waves.
// ---------------------------------------------------------------------------
template <int FA>
__global__ __launch_bounds__(128)
void soi_fused_kernel(const float* __restrict__ features,
                      const int*   __restrict__ ctx,       // context_proposals [F,B,S,2]
                      const float* __restrict__ conv_w,    // [64,256]
                      const float* __restrict__ conv_b,    // [64]
                      const float* __restrict__ fc1_w,     // [64,64]
                      const float* __restrict__ fc1_b,     // [64]
                      const float* __restrict__ fc2_w,     // [20,64]
                      const float* __restrict__ fc2_b,     // [20]
                      float*       __restrict__ scores)    // [F*B*S, 20]
{
    constexpr int L      = 2 << FA;        // 2,4,8,16
    constexpr int G      = L + 1;          // pooling group (torch always pads 256)
    constexpr int repLen = L << 8;         // L*256
    constexpr int CH     = 64 * G;         // dwords per 64-output chunk

    __shared__ float convW[64 * 257];      // padded odd stride -> conflict-free WMMA feeds
    __shared__ float fc1W[64 * 65];
    __shared__ float fc2W[32 * 65];        // rows 20..31 zero
    __shared__ float poolT[16 * 257];
    __shared__ float embT[16 * 65];
    __shared__ float hT[16 * 65];
    __shared__ __align__(16) float spanS[4 * 2 * CH];   // per-wave double-buffered span

    const int tid  = threadIdx.x;
    const int lane = tid & 31;
    const int wv   = tid >> 5;             // 0..3

    const int bx   = blockIdx.x;           // 0..B*(S/16)-1
    const int tile = bx & 63;
    const int b    = bx >> 6;
    const int s0   = tile << 4;

    // ---- stage weights into LDS ----
    for (int idx = tid; idx < 64 * 256; idx += 128)
        convW[(idx >> 8) * 257 + (idx & 255)] = conv_w[idx];
    for (int idx = tid; idx < 64 * 64; idx += 128)
        fc1W[(idx >> 6) * 65 + (idx & 63)] = fc1_w[idx];
    for (int idx = tid; idx < 32 * 65; idx += 128)
        fc2W[idx] = 0.0f;
    __syncthreads();
    for (int idx = tid; idx < 20 * 64; idx += 128)
        fc2W[(idx >> 6) * 65 + (idx & 63)] = fc2_w[idx];

    // ---- async-staged gather + grouped max pool ----
    // generic->LDS: low 32 bits of a __shared__ pointer are the LDS byte offset
    const uint32_t span_base = (uint32_t)(uintptr_t)(&spanS[0]);
    const uint32_t slice     = span_base + (uint32_t)(wv * 2 * CH) * 4u;
    const float* fbase = features + (size_t)(FA * Bq + b) * Tq * FEATq;

    for (int pi = 0; pi < 4; ++pi) {
        const int sl = pi * 4 + wv;                        // wave-uniform proposal row
        const int s  = s0 + sl;
        const int t0 = ctx[((size_t)(FA * Bq + b) * Sq + s) * 2];
        const float* g = fbase + (size_t)t0 * FEATq;       // contiguous L*256 window

        issue_chunk<G>(g, 0, slice, lane);                 // prime chunk 0 -> slot 0
        for (int ch = 0; ch < 4; ++ch) {
            if (ch < 3)
                issue_chunk<G>(g, (ch + 1) * CH, slice + (uint32_t)(((ch + 1) & 1) * CH) * 4u, lane);
            if (ch < 3) wait_async<G>();                   // prev chunk done (in-order)
            else        wait_async<0>();
            const int slot = (ch & 1) * CH;
#pragma unroll
            for (int u = 0; u < 2; ++u) {
                const int oc = u * 32 + lane;              // 0..63 within chunk
                const int o  = ch * 64 + oc;               // pooled channel 0..255
                const int kg = o * G;
                float m = -3.4e38f;
#pragma unroll
                for (int j = 0; j < G; ++j) {
                    const float x = (kg + j < repLen) ? spanS[wv * 2 * CH + slot + oc * G + j]
                                                      : 0.0f;   // forced zero-pad
                    m = fmaxf(m, x);
                }
                poolT[sl * 257 + o] = m;
            }
        }
    }
    __syncthreads();

    const int mrow = lane & 15;            // A row / B column in 16x16 tile
    const int koff = (lane >> 4) << 1;     // lanes 16-31 hold K=2,3
    const int half = lane >> 4;
    const int n0   = wv << 4;

    // ---- GEMM1: emb = relu(pooled[16,256] x conv_w^T + b) ----
    v8f acc = {};
    for (int k0 = 0; k0 < 256; k0 += 4) {
        const int k = k0 + koff;
        v2f a, bb;
        a.x  = poolT[mrow * 257 + k];
        a.y  = poolT[mrow * 257 + k + 1];
        bb.x = convW[(n0 + mrow) * 257 + k];
        bb.y = convW[(n0 + mrow) * 257 + k + 1];
        acc = __builtin_amdgcn_wmma_f32_16x16x4_f32(false, a, false, bb,
                                                    (short)0, acc, false, false);
    }
    for (int r = 0; r < 8; ++r) {
        const int M = r + (half << 3);
        const int N = n0 + mrow;
        const float v = acc[r] + conv_b[N];
        embT[M * 65 + N] = v > 0.0f ? v : 0.0f;
    }
    __syncthreads();

    // ---- GEMM2: h = emb[16,64] x fc1_w^T + b (no relu) ----
    v8f acc2 = {};
    for (int k0 = 0; k0 < 64; k0 += 4) {
        const int k = k0 + koff;
        v2f a, bb;
        a.x  = embT[mrow * 65 + k];
        a.y  = embT[mrow * 65 + k + 1];
        bb.x = fc1W[(n0 + mrow) * 65 + k];
        bb.y = fc1W[(n0 + mrow) * 65 + k + 1];
        acc2 = __builtin_amdgcn_wmma_f32_16x16x4_f32(false, a, false, bb,
                                                     (short)0, acc2, false, false);
    }
    for (int r = 0; r < 8; ++r) {
        const int M = r + (half << 3);
        const int N = n0 + mrow;
        hT[M * 65 + N] = acc2[r] + fc1_b[N];
    }
    __syncthreads();

    // ---- GEMM3: scores = relu(h[16,64] x fc2_w^T + b), N=20 -> 2 tiles ----
    if (wv < 2) {
        const int n0b = wv << 4;
        v8f acc3 = {};
        for (int k0 = 0; k0 < 64; k0 += 4) {
            const int k = k0 + koff;
            v2f a, bb;
            a.x  = hT[mrow * 65 + k];
            a.y  = hT[mrow * 65 + k + 1];
            bb.x = fc2W[(n0b + mrow) * 65 + k];
            bb.y = fc2W[(n0b + mrow) * 65 + k + 1];
            acc3 = __builtin_amdgcn_wmma_f32_16x16x4_f32(false, a, false, bb,
                                                         (short)0, acc3, false, false);
        }
        for (int r = 0; r < 8; ++r) {
            const int M = r + (half << 3);
            const int N = n0b + mrow;
            if (N < Cq) {
                float v = acc3[r] + fc2_b[N];
                v = v > 0.0f ? v : 0.0f;
                scores[((size_t)(FA * Bq + b) * Sq + s0 + M) * Cq + N] = v;
            }
        }
    }
}

// ---------------------------------------------------------------------------
// Scatter-add scores into diff [B,T+1,C] and cdiff [B,T+1]
// ---------------------------------------------------------------------------
__global__ __launch_bounds__(256)
void scatter_kernel(const float* __restrict__ scores,
                    const int*   __restrict__ props,
                    float* __restrict__ diff,
                    float* __restrict__ cdiff)
{
    const int i = blockIdx.x * 256 + threadIdx.x;
    if (i >= Fq * Bq * Sq) return;
    const int b  = (i >> 10) & 15;
    const int st = props[2 * i];
    const int en = props[2 * i + 1];
    float* ds = diff + (size_t)(b * (Tq + 1) + st) * Cq;
    float* de = diff + (size_t)(b * (Tq + 1) + en) * Cq;
    const float* sc = scores + (size_t)i * Cq;
    for (int c = 0; c < Cq; ++c) {
        const float v = sc[c];
        atomicAdd(ds + c, v);
        atomicAdd(de + c, -v);
    }
    atomicAdd(cdiff + b * (Tq + 1) + st, 1.0f);
    atomicAdd(cdiff + b * (Tq + 1) + en, -1.0f);
}

// ---------------------------------------------------------------------------
// counter = cumsum(cdiff) per b (wave32 inclusive scan)
// ---------------------------------------------------------------------------
__global__ __launch_bounds__(32)
void counter_scan_kernel(const float* __restrict__ cdiff, float* __restrict__ counter)
{
    const int b = blockIdx.x;
    const int lane = threadIdx.x;
    float carry = 0.0f;
    for (int ch = 0; ch < Tq / 32; ++ch) {
        const int t = ch * 32 + lane;
        float x = cdiff[b * (Tq + 1) + t];
        for (int d = 1; d < 32; d <<= 1) {
            float y = __shfl_up(x, d);
            if (lane >= d) x += y;
        }
        x += carry;
        counter[b * Tq + t] = x;
        carry = __shfl(x, 31);
    }
}

// ---------------------------------------------------------------------------
// cas = cumsum(diff)/max(counter,1); accumulate actioness
// ---------------------------------------------------------------------------
__global__ __launch_bounds__(32)
void cas_scan_kernel(const float* __restrict__ diff,
                     const float* __restrict__ counter,
                     float* __restrict__ out_cas,
                     float* __restrict__ out_act)
{
    const int g = blockIdx.x;
    const int b = g / Cq;
    const int c = g % Cq;
    const int lane = threadIdx.x;
    float carry = 0.0f;
    for (int ch = 0; ch < Tq / 32; ++ch) {
        const int t = ch * 32 + lane;
        float x = diff[(size_t)(b * (Tq + 1) + t) * Cq + c];
        for (int d = 1; d < 32; d <<= 1) {
            float y = __shfl_up(x, d);
            if (lane >= d) x += y;
        }
        x += carry;
        float cnt = counter[b * Tq + t];
        if (cnt == 0.0f) cnt = 1.0f;
        const float cas = x / cnt;
        out_cas[(size_t)(b * Tq + t) * Cq + c] = cas;
        atomicAdd(out_act + b * Tq + t, cas);
        carry = __shfl(x, 31);
    }
}

// ---------------------------------------------------------------------------
// per (b,c): bitonic sort of cas over T, mean of top-819
// ---------------------------------------------------------------------------
__global__ __launch_bounds__(256)
void topk_kernel(const float* __restrict__ out_cas, float* __restrict__ means)
{
    __shared__ float v[Tq];
    __shared__ float red[256];
    const int b = blockIdx.x / Cq;
    const int c = blockIdx.x % Cq;
    const int tid = threadIdx.x;

    for (int i = tid; i < Tq; i += 256)
        v[i] = out_cas[(size_t)(b * Tq + i) * Cq + c];
    __syncthreads();

    for (int k = 2; k <= Tq; k <<= 1) {
        for (int j = k >> 1; j > 0; j >>= 1) {
            for (int i = tid; i < Tq; i += 256) {
                const int x = i ^ j;
                if (x > i) {
                    const float a = v[i], bb = v[x];
                    const bool sw = ((i & k) == 0) ? (a > bb) : (a < bb);
                    if (sw) { v[i] = bb; v[x] = a; }
                }
            }
            __syncthreads();
        }
    }
    float s = 0.0f;
    for (int i = NDROP + tid; i < Tq; i += 256) s += v[i];
    red[tid] = s;
    __syncthreads();
    for (int w = 128; w > 0; w >>= 1) {
        if (tid < w) red[tid] += red[tid + w];
        __syncthreads();
    }
    if (tid == 0) means[b * Cq + c] = red[0] / (float)KEASY;
}

// ---------------------------------------------------------------------------
// softmax over classes -> video_scores
// ---------------------------------------------------------------------------
__global__ __launch_bounds__(32)
void softmax_kernel(const float* __restrict__ means, float* __restrict__ vs)
{
    const int b = blockIdx.x;
    const int lane = threadIdx.x;
    const float x = (lane < Cq) ? means[b * Cq + lane] : -3.4e38f;
    float m = x;
    for (int d = 16; d > 0; d >>= 1) m = fmaxf(m, __shfl_xor(m, d));
    const float e = (lane < Cq) ? __expf(x - m) : 0.0f;
    float s = e;
    for (int d = 16; d > 0; d >>= 1) s += __shfl_xor(s, d);
    if (lane < Cq) vs[b * Cq + lane] = e / s;
}

// ---------------------------------------------------------------------------
extern "C" void kernel_launch(void* const* d_in, const int* in_sizes, int n_in,
                              void* d_out, int out_size, void* d_ws, size_t ws_size,
                              hipStream_t stream)
{
    (void)in_sizes; (void)n_in; (void)out_size; (void)ws_size;
    const float* features = (const float*)d_in[0];
    const int*   props    = (const int*)d_in[1];
    const int*   ctx      = (const int*)d_in[2];
    const float* conv_w   = (const float*)d_in[3];
    const float* conv_b   = (const float*)d_in[4];
    const float* fc1_w    = (const float*)d_in[5];
    const float* fc1_b    = (const float*)d_in[6];
    const float* fc2_w    = (const float*)d_in[7];
    const float* fc2_b    = (const float*)d_in[8];

    float* sc      = (float*)d_ws;                       // F*B*S*20
    float* diff    = sc + (size_t)Fq * Bq * Sq * Cq;     // B*(T+1)*C
    float* cdiff   = diff + (size_t)Bq * (Tq + 1) * Cq;  // B*(T+1)
    float* counter = cdiff + (size_t)Bq * (Tq + 1);      // B*T
    float* means   = counter + (size_t)Bq * Tq;          // B*C

    float* out   = (float*)d_out;
    float* o_vs  = out;
    float* o_act = out + Bq * Cq;
    float* o_cas = o_act + Bq * Tq;

    hipMemsetAsync(diff, 0, ((size_t)Bq * (Tq + 1) * Cq + (size_t)Bq * (Tq + 1)) * sizeof(float), stream);
    hipMemsetAsync(o_act, 0, (size_t)Bq * Tq * sizeof(float), stream);

    const int grid = Bq * (Sq / 16);
    soi_fused_kernel<0><<<grid, 128, 0, stream>>>(features, ctx, conv_w, conv_b, fc1_w, fc1_b, fc2_w, fc2_b, sc);
    soi_fused_kernel<1><<<grid, 128, 0, stream>>>(features, ctx, conv_w, conv_b, fc1_w, fc1_b, fc2_w, fc2_b, sc);
    soi_fused_kernel<2><<<grid, 128, 0, stream>>>(features, ctx, conv_w, conv_b, fc1_w, fc1_b, fc2_w, fc2_b, sc);
    soi_fused_kernel<3><<<grid, 128, 0, stream>>>(features, ctx, conv_w, conv_b, fc1_w, fc1_b, fc2_w, fc2_b, sc);

    scatter_kernel<<<(Fq * Bq * Sq) / 256, 256, 0, stream>>>(sc, props, diff, cdiff);
    counter_scan_kernel<<<Bq, 32, 0, stream>>>(cdiff, counter);
    cas_scan_kernel<<<Bq * Cq, 32, 0, stream>>>(diff, counter, o_cas, o_act);
    topk_kernel<<<Bq * Cq, 256, 0, stream>>>(o_cas, means);
    softmax_kernel<<<Bq, 32, 0, stream>>>(means, o_vs);
}